// GTLayer_76647986364551
// MI455X (gfx1250) — compile-verified
//
#include <hip/hip_runtime.h>
#include <math.h>

// Graph-transformer layer for MI455X (gfx1250, wave32, WMMA).
// Roofline: e is 537 MB fp32 -> ~23us floor at 23.3 TB/s. The fused
// edge-projection (M=512,K=128,N=16 per (b,i)) needs ~187 TFLOP/s to stay
// bandwidth-bound -> runs on V_WMMA_F32_16X16X4_F32. e1/e2 live only in LDS.

#define D   128
#define H   8
#define NN  512
#define FF  512
#define EPSV 1e-5f

typedef float v2f __attribute__((ext_vector_type(2)));
typedef float v8f __attribute__((ext_vector_type(8)));

__device__ __forceinline__ float wave_sum(float x) {
    #pragma unroll
    for (int off = 16; off > 0; off >>= 1) x += __shfl_xor(x, off, 32);
    return x;
}
__device__ __forceinline__ float wave_max(float x) {
    #pragma unroll
    for (int off = 16; off > 0; off >>= 1) x = fmaxf(x, __shfl_xor(x, off, 32));
    return x;
}

// ---------------- Kernel A: hn = rmsnorm(h); qkv = hn @ W_h ----------------
// grid = B*N rows, 128 threads (4 waves). W_h (128x384) is L2-resident.
__global__ void qkv_kernel(const float* __restrict__ h, const float* __restrict__ Wh,
                           const float* __restrict__ n1w,
                           float* __restrict__ q, float* __restrict__ k, float* __restrict__ v)
{
    const int row = blockIdx.x;
    const int t   = threadIdx.x;           // 0..127
    __shared__ float hn[D];
    __shared__ float red[4];

    float x  = h[(size_t)row * D + t];
    float ss = wave_sum(x * x);
    if ((t & 31) == 0) red[t >> 5] = ss;
    __syncthreads();
    float sum = red[0] + red[1] + red[2] + red[3];
    float r   = rsqrtf(sum * (1.0f / D) + EPSV);
    hn[t] = x * r * n1w[t];
    __syncthreads();

    float a0 = 0.f, a1 = 0.f, a2 = 0.f;
    #pragma unroll 4
    for (int c = 0; c < D; ++c) {
        const float  z  = hn[c];
        const float* wr = Wh + (size_t)c * (3 * D);   // coalesced across t
        a0 += z * wr[t];
        a1 += z * wr[t + D];
        a2 += z * wr[t + 2 * D];
    }
    q[(size_t)row * D + t] = a0;
    k[(size_t)row * D + t] = a1;
    v[(size_t)row * D + t] = a2;
}

// ------- Kernel B: fused edge-projection (WMMA f32) + biased gated attention -------
// One block per (b,i). 256 threads = 8 waves. Each wave WMMAs 64 rows of
// e[b,i,:,:] @ W_e (M=512,K=128,N=16) into LDS, then scores/softmax/AV.
__global__ void __launch_bounds__(256, 1)
attn_kernel(const float* __restrict__ e, const float* __restrict__ We,
            const float* __restrict__ q, const float* __restrict__ k,
            const float* __restrict__ v, float* __restrict__ y)
{
    const int row  = blockIdx.x;          // b*N + i
    const int b    = row >> 9;            // N = 512
    const int t    = threadIdx.x;
    const int lane = t & 31;
    const int w    = t >> 5;              // wave 0..7

    __shared__ float sWe[D * 16];         // 8 KB   W_e staged for B-operand reads
    __shared__ float sQ[D];               // q row
    __shared__ float sE[NN * 16];         // 32 KB  e12 -> scores(h<8) & gate(h>=8)
    __shared__ float sPart[D];

    for (int idx = t; idx < D * 16; idx += 256) sWe[idx] = We[idx];
    if (t < D) sQ[t] = q[(size_t)row * D + t];
    __syncthreads();

    // ---- WMMA edge projection ----
    // A (f32 16x4): lanes 0-15 -> rows M0..15 with K={4k,4k+1} in v0/v1,
    //               lanes 16-31 -> same rows with K={4k+2,4k+3}. One b64/lane/step.
    const float* erow  = e + (size_t)row * NN * D;
    const int    mrow  = lane & 15;       // M row within tile / N col for B & D
    const int    khalf = lane >> 4;       // 0: K 0,1   1: K 2,3
    const float* ap0 = erow + (size_t)(w * 64 +  0 + mrow) * D + khalf * 2;
    const float* ap1 = ap0 + 16 * D;
    const float* ap2 = ap0 + 32 * D;
    const float* ap3 = ap0 + 48 * D;
    const int    wb  = (khalf * 2) * 16 + mrow;   // B: W_e[K-row][N-col], K pair per half

    v8f acc0 = {}, acc1 = {}, acc2 = {}, acc3 = {};
    for (int kk = 0; kk < 32; ++kk) {
        v2f Bv;
        Bv.x = sWe[wb + kk * 64];
        Bv.y = sWe[wb + kk * 64 + 16];
        v2f A0 = *(const v2f*)(ap0 + kk * 4);
        v2f A1 = *(const v2f*)(ap1 + kk * 4);
        v2f A2 = *(const v2f*)(ap2 + kk * 4);
        v2f A3 = *(const v2f*)(ap3 + kk * 4);
        acc0 = __builtin_amdgcn_wmma_f32_16x16x4_f32(false, A0, false, Bv, (short)0, acc0, false, false);
        acc1 = __builtin_amdgcn_wmma_f32_16x16x4_f32(false, A1, false, Bv, (short)0, acc1, false, false);
        acc2 = __builtin_amdgcn_wmma_f32_16x16x4_f32(false, A2, false, Bv, (short)0, acc2, false, false);
        acc3 = __builtin_amdgcn_wmma_f32_16x16x4_f32(false, A3, false, Bv, (short)0, acc3, false, false);
    }
    // D layout: lane L, VGPR r -> (M = r + 8*(L>>4), N = L&15)
    #pragma unroll
    for (int r = 0; r < 8; ++r) {
        const int jj = w * 64 + r + 8 * khalf;
        sE[(jj +  0) * 16 + mrow] = acc0[r];
        sE[(jj + 16) * 16 + mrow] = acc1[r];
        sE[(jj + 32) * 16 + mrow] = acc2[r];
        sE[(jj + 48) * 16 + mrow] = acc3[r];
    }
    __syncthreads();

    // ---- scores: a[h][j] = q_h . k_jh / sqrt(16) + e1[h][j]  (k rows are L2-hot)
    const float* kb = k + (size_t)b * NN * D;
    for (int jj = t; jj < NN; jj += 256) {
        const float* kr = kb + (size_t)jj * D;
        #pragma unroll
        for (int hh = 0; hh < H; ++hh) {
            float s = 0.f;
            #pragma unroll
            for (int d2 = 0; d2 < 16; ++d2) s += sQ[hh * 16 + d2] * kr[hh * 16 + d2];
            sE[jj * 16 + hh] = s * 0.25f + sE[jj * 16 + hh];
        }
    }
    __syncthreads();

    // ---- per-head softmax * e2: wave w owns head h=w (8 waves == 8 heads)
    {
        const int hh = w;
        float m = -1e30f;
        for (int jj = lane; jj < NN; jj += 32) m = fmaxf(m, sE[jj * 16 + hh]);
        m = wave_max(m);
        float s = 0.f;
        for (int jj = lane; jj < NN; jj += 32) s += __expf(sE[jj * 16 + hh] - m);
        s = wave_sum(s);
        const float inv = 1.0f / s;
        for (int jj = lane; jj < NN; jj += 32)
            sE[jj * 16 + hh] = __expf(sE[jj * 16 + hh] - m) * inv * sE[jj * 16 + 8 + hh];
    }
    __syncthreads();

    // ---- y[h][d] = sum_j s[h][j] * v[b,j,h*16+d]   (v loads coalesced across d)
    const float* vb   = v + (size_t)b * NN * D;
    const int    dcol = t & 127;
    const int    hsel = t >> 7;
    const int    hh2  = dcol >> 4;
    float acc = 0.f;
    for (int jj = hsel * 256; jj < hsel * 256 + 256; ++jj)
        acc += sE[jj * 16 + hh2] * vb[(size_t)jj * D + dcol];
    if (hsel) sPart[dcol] = acc;
    __syncthreads();
    if (!hsel) y[(size_t)row * D + dcol] = acc + sPart[dcol];
}

// -------- Kernel C: z = rmsnorm(y+h); out = gelu(z@W_fc)@W_proj + y --------
// grid = B*N rows, 256 threads. W_fc/W_proj L2-resident.
__global__ void mlp_kernel(const float* __restrict__ hin, const float* __restrict__ y,
                           const float* __restrict__ n2w,
                           const float* __restrict__ Wfc, const float* __restrict__ Wp,
                           float* __restrict__ out)
{
    const int row = blockIdx.x;
    const int t   = threadIdx.x;          // 0..255
    __shared__ float z[D];
    __shared__ float g[FF];
    __shared__ float red[8];
    __shared__ float part[D];

    float x = 0.f;
    if (t < D) x = y[(size_t)row * D + t] + hin[(size_t)row * D + t];
    float ss = wave_sum(x * x);
    if ((t & 31) == 0) red[t >> 5] = ss;
    __syncthreads();
    float sum = 0.f;
    #pragma unroll
    for (int i = 0; i < 8; ++i) sum += red[i];
    const float r = rsqrtf(sum * (1.0f / D) + EPSV);
    if (t < D) z[t] = x * r * n2w[t];
    __syncthreads();

    #pragma unroll
    for (int o = 0; o < 2; ++o) {
        const int f = t + o * 256;
        float a = 0.f;
        #pragma unroll 4
        for (int c = 0; c < D; ++c) a += z[c] * Wfc[(size_t)c * FF + f];  // coalesced
        g[f] = 0.5f * a * (1.0f + erff(a * 0.70710678118f));              // exact gelu
    }
    __syncthreads();

    const int dcol = t & 127, hsel = t >> 7;
    float a = 0.f;
    for (int f = hsel * 256; f < hsel * 256 + 256; ++f)
        a += g[f] * Wp[(size_t)f * D + dcol];                             // coalesced
    if (hsel) part[dcol] = a;
    __syncthreads();
    if (!hsel)
        out[(size_t)row * D + dcol] = a + part[dcol] + y[(size_t)row * D + dcol];
}

extern "C" void kernel_launch(void* const* d_in, const int* in_sizes, int n_in,
                              void* d_out, int out_size, void* d_ws, size_t ws_size,
                              hipStream_t stream)
{
    const float* h   = (const float*)d_in[0];
    const float* e   = (const float*)d_in[1];
    const float* Wh  = (const float*)d_in[2];
    const float* We  = (const float*)d_in[3];
    const float* n1w = (const float*)d_in[4];
    const float* n2w = (const float*)d_in[5];
    const float* Wfc = (const float*)d_in[6];
    const float* Wp  = (const float*)d_in[7];
    float* out = (float*)d_out;

    const int rows = in_sizes[0] / D;     // B*N = 2048
    float* ws = (float*)d_ws;             // q,k,v,y : 4 x rows*D floats (4 MB)
    float* q = ws;
    float* k = q + (size_t)rows * D;
    float* v = k + (size_t)rows * D;
    float* y = v + (size_t)rows * D;

    qkv_kernel <<<rows, 128, 0, stream>>>(h, Wh, n1w, q, k, v);
    attn_kernel<<<rows, 256, 0, stream>>>(e, We, q, k, v, y);
    mlp_kernel <<<rows, 256, 0, stream>>>(h, y, n2w, Wfc, Wp, out);
}